// ProgressiveFocusedAttention_15745350107553
// MI455X (gfx1250) — compile-verified
//
#include <hip/hip_runtime.h>
#include <hip/hip_bf16.h>
#include <math.h>

typedef __attribute__((ext_vector_type(2))) float v2f;
typedef __attribute__((ext_vector_type(8))) float v8f;

#define B_    128
#define H_    6
#define N_    256
#define K_IN  192
#define HD_   32
#define C_    192      // H_*HD_
#define FOURC 768      // 4*C_
#define TOPK_ 128
#define SCALE_ 0.17677669529663687f  // 32^-0.5
#define EPS_  1e-20f

__device__ __forceinline__ float wave_max32(float v) {
#pragma unroll
  for (int s = 16; s > 0; s >>= 1) v = fmaxf(v, __shfl_xor(v, s, 32));
  return v;
}
__device__ __forceinline__ float wave_sum32(float v) {
#pragma unroll
  for (int s = 16; s > 0; s >>= 1) v += __shfl_xor(v, s, 32);
  return v;
}

// One block per (b, h, 16-row tile). 256 threads = 8 wave32.
__launch_bounds__(256)
__global__ void pfa_attn_kernel(const float* __restrict__ qkvp,
                                const float* __restrict__ pfa_values,
                                const int*   __restrict__ pfa_indices,
                                const int*   __restrict__ rpi,
                                const float* __restrict__ rpb,
                                float* __restrict__ x_ws,       // (B,N,C)
                                float* __restrict__ out_vals,   // (B,H,N,TOPK)
                                float* __restrict__ out_idx) {  // (B,H,N,TOPK) as float
  const int blk  = blockIdx.x;       // B_*H_*16 blocks
  const int rt   = blk & 15;         // row tile within (b,h)
  const int bh   = blk >> 4;
  const int h    = bh % H_;
  const int b    = bh / H_;
  const int row0 = rt * 16;

  const int tid  = threadIdx.x;
  const int lane = tid & 31;
  const int wave = tid >> 5;

  __shared__ float logits_s[16][N_];     // 16 KB
  __shared__ float attn_s[8][K_IN];      // 6 KB (per-wave scratch)
  __shared__ int   midx_s[8][K_IN];      // 6 KB

  // ---------- QK^T (16 x 256) via V_WMMA_F32_16X16X4_F32 ----------
  // A frag (16x4 f32): lanes 0-15 -> M=lane, K={0,1}; lanes 16-31 -> M=lane-16, K={2,3}
  const int mrow = lane & 15;
  const int koff = (lane >> 4) * 2;
  const long long qbase = (long long)(b * N_ + row0 + mrow) * FOURC + h * HD_;       // q slot
#pragma unroll
  for (int w = 0; w < 2; ++w) {
    const int nt   = wave * 2 + w;                 // column tile 0..15
    const int ncol = nt * 16 + mrow;               // K-matrix row = logits column
    const long long kbase = (long long)(b * N_ + ncol) * FOURC + C_ + h * HD_;       // k slot
    v8f acc = {};
#pragma unroll
    for (int k = 0; k < HD_; k += 4) {
      v2f a = *(const v2f*)(qkvp + qbase + k + koff);
      a *= SCALE_;                                  // fold softmax scale into A
      v2f bt = *(const v2f*)(qkvp + kbase + k + koff); // B = K^T frag
      acc = __builtin_amdgcn_wmma_f32_16x16x4_f32(false, a, false, bt,
                                                  (short)0, acc, false, false);
    }
    // D layout: VGPR r -> M = r + 8*(lane>>4), N = lane&15
#pragma unroll
    for (int r = 0; r < 8; ++r) {
      const int m = r + 8 * (lane >> 4);
      logits_s[m][nt * 16 + mrow] = acc[r];
    }
  }
  __syncthreads();

  // ---------- per-row: gather + bias, softmax, modulate, renorm, top-k, A@V ----------
  // uniform trip count: every wave does rr = wave, wave+8
  for (int rr = wave; rr < 16; rr += 8) {
    const int n = row0 + rr;
    const long long rowbase = ((long long)bh * N_ + n) * K_IN;

    float myv[6]; int myj[6]; int mym[6];
    float vmax = -3.4e38f;
#pragma unroll
    for (int t = 0; t < 6; ++t) {
      const int j  = lane + 32 * t;
      const int mi = pfa_indices[rowbase + j];
      const int rp = rpi[n * N_ + mi];
      const float lg = logits_s[rr][mi] + rpb[rp * H_ + h];
      myv[t] = lg; myj[t] = j; mym[t] = mi;
      vmax = fmaxf(vmax, lg);
    }
    vmax = wave_max32(vmax);

    float esum = 0.f;
#pragma unroll
    for (int t = 0; t < 6; ++t) { myv[t] = expf(myv[t] - vmax); esum += myv[t]; }
    esum = wave_sum32(esum);
    const float inv = 1.0f / esum;

    float msum = 0.f;
#pragma unroll
    for (int t = 0; t < 6; ++t) {
      myv[t] = myv[t] * inv * pfa_values[rowbase + myj[t]];   // progressive focusing
      msum += myv[t];
    }
    msum = wave_sum32(msum);
    const float rinv = 1.0f / (msum + EPS_);
#pragma unroll
    for (int t = 0; t < 6; ++t) {
      myv[t] = (myv[t] + EPS_) * rinv;
      attn_s[wave][myj[t]] = myv[t];
      midx_s[wave][myj[t]] = mym[t];
    }
    __syncthreads();

    // exact top-128, sorted descending, stable tie-break (lowest j first) == jax top_k
    int rank[6] = {0, 0, 0, 0, 0, 0};
    for (int u = 0; u < K_IN; ++u) {
      const float au = attn_s[wave][u];
#pragma unroll
      for (int t = 0; t < 6; ++t)
        rank[t] += (au > myv[t]) || (au == myv[t] && u < myj[t]);
    }
    const long long ob = ((long long)bh * N_ + n) * TOPK_;
    bool keep[6];
#pragma unroll
    for (int t = 0; t < 6; ++t) {
      keep[t] = rank[t] < TOPK_;
      if (keep[t]) {
        out_vals[ob + rank[t]] = myv[t];
        out_idx [ob + rank[t]] = (float)mym[t];
      }
    }
    __syncthreads();
#pragma unroll
    for (int t = 0; t < 6; ++t) attn_s[wave][myj[t]] = keep[t] ? myv[t] : 0.f;
    __syncthreads();

    // x[n][d] = sum_j w_j * v[m_j][d] + v_lepe[n][d]   (lane == d, HD_ == 32)
    const long long nb = (long long)(b * N_ + n) * FOURC + h * HD_ + lane;
    float acc = qkvp[nb + 3 * C_];                 // v_lepe
    for (int u = 0; u < K_IN; ++u) {
      const float wgt = attn_s[wave][u];
      const int   m   = midx_s[wave][u];
      acc += wgt * qkvp[(long long)(b * N_ + m) * FOURC + 2 * C_ + h * HD_ + lane];
    }
    x_ws[(long long)(b * N_ + n) * C_ + h * HD_ + lane] = acc;
    __syncthreads();
  }
}

// Projection: out(32768 x 192) = x @ W^T + b, fp32 WMMA, one 16x16 tile per wave.
__launch_bounds__(256)
__global__ void pfa_proj_kernel(const float* __restrict__ x,
                                const float* __restrict__ w,     // (C,C) row-major
                                const float* __restrict__ bias,  // (C)
                                float* __restrict__ out) {
  const int tid  = threadIdx.x;
  const int lane = tid & 31;
  const int wave = tid >> 5;
  const int tile = blockIdx.x * 8 + wave;        // 2048*12 tiles total
  const int mt   = tile / 12;
  const int nt   = tile % 12;

  const int mrow = lane & 15;
  const int kb   = (lane >> 4) * 2;
  const float* arow = x + (long long)(mt * 16 + mrow) * C_;    // A: x rows
  const float* brow = w + (long long)(nt * 16 + mrow) * C_;    // B = W^T: columns of out
  v8f acc = {};
#pragma unroll
  for (int k = 0; k < C_; k += 4) {
    v2f a  = *(const v2f*)(arow + k + kb);
    v2f bt = *(const v2f*)(brow + k + kb);
    acc = __builtin_amdgcn_wmma_f32_16x16x4_f32(false, a, false, bt,
                                                (short)0, acc, false, false);
  }
  const int nc  = nt * 16 + mrow;                // N = lane&15
  const float pb = bias[nc];
#pragma unroll
  for (int r = 0; r < 8; ++r) {
    const int m = r + 8 * (lane >> 4);
    out[(long long)(mt * 16 + m) * C_ + nc] = acc[r] + pb;
  }
}

extern "C" void kernel_launch(void* const* d_in, const int* in_sizes, int n_in,
                              void* d_out, int out_size, void* d_ws, size_t ws_size,
                              hipStream_t stream) {
  (void)in_sizes; (void)n_in; (void)out_size; (void)ws_size;
  const float* qkvp        = (const float*)d_in[0];
  const float* pfa_values  = (const float*)d_in[1];
  const int*   pfa_indices = (const int*)  d_in[2];
  const int*   rpi         = (const int*)  d_in[3];
  const float* rpb_table   = (const float*)d_in[4];
  const float* proj_w      = (const float*)d_in[5];
  const float* proj_b      = (const float*)d_in[6];

  float* out      = (float*)d_out;
  float* out_x    = out;                                        // B*N*C      = 6,291,456
  float* out_vals = out_x + (size_t)B_ * N_ * C_;               // B*H*N*TOPK = 25,165,824
  float* out_idx  = out_vals + (size_t)B_ * H_ * N_ * TOPK_;
  float* x_ws     = (float*)d_ws;                               // B*N*C floats

  pfa_attn_kernel<<<B_ * H_ * (N_ / 16), 256, 0, stream>>>(
      qkvp, pfa_values, pfa_indices, rpi, rpb_table, x_ws, out_vals, out_idx);

  const int mtiles = (B_ * N_) / 16;   // 2048
  const int ntiles = C_ / 16;          // 12
  pfa_proj_kernel<<<(mtiles * ntiles) / 8, 256, 0, stream>>>(
      x_ws, proj_w, proj_b, out_x);
}